// CantorMultiheadFusion_75668733821280
// MI455X (gfx1250) — compile-verified
//
#include <hip/hip_runtime.h>
#include <hip/hip_bf16.h>

typedef float v2f __attribute__((ext_vector_type(2)));
typedef float v8f __attribute__((ext_vector_type(8)));
typedef int   v4i __attribute__((ext_vector_type(4)));

#define TM  128   // block tile M
#define TNB 128   // block tile N
#define TK  32    // K tile
#define LDA 36    // 32 + 4 pad: 16B-aligned rows, conflict-free strided reads

// ---- CDNA5 async global->LDS copy (ASYNCcnt path), with sync fallback ----
#if defined(__has_builtin)
#if __has_builtin(__builtin_amdgcn_global_load_async_to_lds_b128)
#define HAVE_ASYNC_LDS 1
#endif
#endif

typedef __attribute__((address_space(1))) v4i gv4i;   // global int4
typedef __attribute__((address_space(3))) v4i lv4i;   // LDS int4

__device__ __forceinline__ void copy16_g2l(const float* gp, float* lp) {
#if defined(HAVE_ASYNC_LDS)
    __builtin_amdgcn_global_load_async_to_lds_b128(
        (gv4i*)gp, (lv4i*)lp, /*offset=*/0, /*cpol=*/0);
#else
    *(float4*)lp = *(const float4*)gp;
#endif
}

__device__ __forceinline__ void wait_async_copies() {
#if defined(HAVE_ASYNC_LDS)
#if __has_builtin(__builtin_amdgcn_s_wait_asynccnt)
    __builtin_amdgcn_s_wait_asynccnt(0);
#else
    asm volatile("s_wait_asynccnt 0" ::: "memory");
#endif
#endif
}

// Stage a 128-row x 32-col f32 tile (row-major, source stride Kdim) into LDS.
// 1024 x 16B chunks, 4 per thread (256 threads).
__device__ __forceinline__ void stage_tile(const float* __restrict__ src,
                                           float* __restrict__ dst,
                                           int row0, int Kdim, int k0, int tid) {
    #pragma unroll
    for (int i = 0; i < 4; ++i) {
        int idx = tid + i * 256;
        int r   = idx >> 3;  // 0..127
        int c4  = idx & 7;   // 0..7
        copy16_g2l(&src[(size_t)(row0 + r) * Kdim + k0 + c4 * 4],
                   &dst[r * LDA + c4 * 4]);
    }
}

// C[M,N] = A[M,K] * Wt[N,K]^T  (+ bias[col] + resid[row,col] if provided)
// 8 waves: 4 along M x 2 along N; wave tile 32x64 = 2x4 WMMA tiles.
__global__ __launch_bounds__(256) void gemm_f32_wmma(
    const float* __restrict__ A,     // M x Kdim row-major
    const float* __restrict__ Wt,    // N x Kdim row-major (B transposed)
    float* __restrict__ C,           // M x N
    const float* __restrict__ bias,  // nullptr or [N]
    const float* __restrict__ resid, // nullptr or [M,N]
    int M, int N, int Kdim)
{
    // double-buffered tiles, both kept in natural row-major (no transpose):
    // fragment (k,k+1) pairs are contiguous -> single ds_load_b64 each.
    __shared__ float As[2][TM  * LDA];   // 2 x 18KB
    __shared__ float Bs[2][TNB * LDA];   // 2 x 18KB

    const int tid  = threadIdx.x;
    const int lane = tid & 31;
    const int wid  = tid >> 5;   // 0..7
    const int wm   = wid & 3;    // 4 waves along M  -> 32 rows each
    const int wn   = wid >> 2;   // 2 waves along N  -> 64 cols each
    const int blockM = blockIdx.y * TM;
    const int blockN = blockIdx.x * TNB;

    const int lcol  = lane & 15;         // row/col within 16-wide tile
    const int khalf = (lane >> 4) << 1;  // lanes 16..31 carry K+2 (f32 16x16x4 layout)

    v8f acc[2][4] = {};

    int buf = 0;
    stage_tile(A,  As[0], blockM, Kdim, 0, tid);
    stage_tile(Wt, Bs[0], blockN, Kdim, 0, tid);
    wait_async_copies();
    __syncthreads();

    for (int k0 = 0; k0 < Kdim; k0 += TK) {
        // prefetch next K tile into the other buffer (overlaps with WMMAs)
        if (k0 + TK < Kdim) {
            stage_tile(A,  As[buf ^ 1], blockM, Kdim, k0 + TK, tid);
            stage_tile(Wt, Bs[buf ^ 1], blockN, Kdim, k0 + TK, tid);
        }

        const float* as = &As[buf][0];
        const float* bs = &Bs[buf][0];
        #pragma unroll
        for (int kk = 0; kk < TK; kk += 4) {
            v2f a[2], b[4];
            #pragma unroll
            for (int mt = 0; mt < 2; ++mt)
                a[mt] = *(const v2f*)&as[(wm * 32 + mt * 16 + lcol) * LDA + kk + khalf];
            #pragma unroll
            for (int nt = 0; nt < 4; ++nt)
                b[nt] = *(const v2f*)&bs[(wn * 64 + nt * 16 + lcol) * LDA + kk + khalf];
            #pragma unroll
            for (int mt = 0; mt < 2; ++mt)
                #pragma unroll
                for (int nt = 0; nt < 4; ++nt)
                    acc[mt][nt] = __builtin_amdgcn_wmma_f32_16x16x4_f32(
                        false, a[mt], false, b[nt],
                        (short)0, acc[mt][nt], false, false);
        }

        wait_async_copies();
        __syncthreads();
        buf ^= 1;
    }

    // epilogue: VGPR j -> M = j (lanes 0-15) or j+8 (lanes 16-31); N = lane&15
    const int rhalf = (lane >> 4) << 3;
    #pragma unroll
    for (int mt = 0; mt < 2; ++mt) {
        #pragma unroll
        for (int nt = 0; nt < 4; ++nt) {
            int col = blockN + wn * 64 + nt * 16 + lcol;
            #pragma unroll
            for (int j = 0; j < 8; ++j) {
                int row = blockM + wm * 32 + mt * 16 + rhalf + j;
                float v = acc[mt][nt][j];
                if (bias)  v += bias[col];
                if (resid) v += resid[(size_t)row * N + col];
                C[(size_t)row * N + col] = v;
            }
        }
    }
}

// w[s,k] = softmax_k(-distances[s, routes[s,k]]) ; one wave per row, K=64
__global__ __launch_bounds__(256) void softmax_routes(
    const int* __restrict__ routes, const float* __restrict__ dist,
    float* __restrict__ w, int S, int Kn)
{
    int wid  = threadIdx.x >> 5;
    int lane = threadIdx.x & 31;
    int s = blockIdx.x * 8 + wid;
    if (s >= S) return;

    int r0 = routes[s * Kn + lane];
    int r1 = routes[s * Kn + lane + 32];
    float v0 = -dist[(size_t)s * S + r0];
    float v1 = -dist[(size_t)s * S + r1];

    float m = fmaxf(v0, v1);
    #pragma unroll
    for (int off = 16; off >= 1; off >>= 1)
        m = fmaxf(m, __shfl_xor(m, off, 32));

    float e0 = __expf(v0 - m);
    float e1 = __expf(v1 - m);
    float sum = e0 + e1;
    #pragma unroll
    for (int off = 16; off >= 1; off >>= 1)
        sum += __shfl_xor(sum, off, 32);

    float inv = 1.0f / sum;
    w[s * Kn + lane]      = e0 * inv;
    w[s * Kn + lane + 32] = e1 * inv;
}

// fused[b,s,:] = sum_k w[s,k] * h[b, routes[s,k], :]
__global__ __launch_bounds__(128) void fuse_gather(
    const float* __restrict__ h, const int* __restrict__ routes,
    const float* __restrict__ w, float* __restrict__ fused,
    int S, int D, int Kn)
{
    __shared__ int   r_sh[64];
    __shared__ float w_sh[64];

    int bs = blockIdx.x;   // 0 .. B*S-1
    int b  = bs >> 11;     // S == 2048
    int s  = bs & 2047;
    int tid = threadIdx.x;

    if (tid < 64) {
        r_sh[tid] = routes[s * Kn + tid];
        w_sh[tid] = w[s * Kn + tid];
    }
    __syncthreads();

    int col = tid * 4;
    const float* hb = h + (size_t)b * S * D;
    float4 acc = make_float4(0.f, 0.f, 0.f, 0.f);
    #pragma unroll 4
    for (int k = 0; k < 64; ++k) {
        float wk = w_sh[k];
        float4 g = *(const float4*)(&hb[(size_t)r_sh[k] * D + col]);
        acc.x = fmaf(wk, g.x, acc.x);
        acc.y = fmaf(wk, g.y, acc.y);
        acc.z = fmaf(wk, g.z, acc.z);
        acc.w = fmaf(wk, g.w, acc.w);
    }
    *(float4*)(&fused[(size_t)bs * D + col]) = acc;
}

extern "C" void kernel_launch(void* const* d_in, const int* in_sizes, int n_in,
                              void* d_out, int out_size, void* d_ws, size_t ws_size,
                              hipStream_t stream) {
    const int B = 2, S = 2048, D = 512, Kn = 64;
    const int M = B * S;   // 4096

    const float* x      = (const float*)d_in[0];
    const int*   routes = (const int*)  d_in[1];
    const float* dist   = (const float*)d_in[2];
    const float* W_in   = (const float*)d_in[3];
    const float* W_out  = (const float*)d_in[4];
    const float* b_out  = (const float*)d_in[5];
    float* out = (float*)d_out;

    float* h     = (float*)d_ws;                 // B*S*D floats
    float* w     = h + (size_t)M * D;            // S*K floats
    float* fused = w + (size_t)S * Kn;           // B*S*D floats

    dim3 ggrid(D / TNB, M / TM);                 // (4, 32)

    gemm_f32_wmma<<<ggrid, 256, 0, stream>>>(x, W_in, h, nullptr, nullptr, M, D, D);
    softmax_routes<<<S / 8, 256, 0, stream>>>(routes, dist, w, S, Kn);
    fuse_gather<<<M, 128, 0, stream>>>(h, routes, w, fused, S, D, Kn);
    gemm_f32_wmma<<<ggrid, 256, 0, stream>>>(fused, W_out, out, b_out, x, M, D, D);
}